// ECELoss_50861002719367
// MI455X (gfx1250) — compile-verified
//
#include <hip/hip_runtime.h>

typedef __attribute__((ext_vector_type(2))) float v2f;
typedef __attribute__((ext_vector_type(8))) float v8f;

#define ECE_NBINS 15
#define ECE_BLOCKS 1024
#define ECE_TPB 256

// ds_swizzle group-of-32: offset = {1'b0, xor[4:0], or[4:0], and[4:0]}
template <int PAT>
__device__ __forceinline__ float swzf(float x) {
  return __uint_as_float(
      (unsigned)__builtin_amdgcn_ds_swizzle((int)__float_as_uint(x), PAT));
}
template <int PAT>
__device__ __forceinline__ int swzi(int x) {
  return __builtin_amdgcn_ds_swizzle(x, PAT);
}

__global__ void ece_zero_ws(float* ws) {
  if (threadIdx.x < 48) ws[threadIdx.x] = 0.0f;
}

__global__ __launch_bounds__(ECE_TPB) void ece_main(const float* __restrict__ probs,
                                                    const int* __restrict__ labels,
                                                    float* __restrict__ ws, int N) {
  __shared__ float hist[48];
  if (threadIdx.x < 48) hist[threadIdx.x] = 0.0f;
  __syncthreads();

  const int lane = threadIdx.x & 31;
  const int gw = blockIdx.x * (ECE_TPB / 32) + (threadIdx.x >> 5);
  const int totalWaves = gridDim.x * (ECE_TPB / 32);
  const int s = lane >> 3;                    // subgroup 0..3, one row each
  const int j = lane & 7;                     // lane within subgroup
  const int rowsPerSweep = totalWaves * 8;    // 8 rows (2KB) per wave per iter
  const int numIter = (N + rowsPerSweep - 1) / rowsPerSweep;  // wave-uniform

  const int m = lane & 15;                    // A-row / B-col index for this lane
  const bool isN0 = (m == 0), isN1 = (m == 1), isN2 = (m == 2);  // hoisted selectors

  v8f acc = {0.f, 0.f, 0.f, 0.f, 0.f, 0.f, 0.f, 0.f};

  // A(16x4) one-hot(bin) x B(4x16) [conf,acc,1] accumulated into D(16x16) f32.
  auto accumulate = [&](float cA, float cB, int qA, int qB) {
    float vaA = ((qA >> 9) & 1) ? 1.0f : 0.0f;
    float vaB = ((qB >> 9) & 1) ? 1.0f : 0.0f;
    float acA = ((qA >> 8) & 1) ? vaA : 0.0f;
    float acB = ((qB >> 8) & 1) ? vaB : 0.0f;
    int binA = qA & 0xff, binB = qB & 0xff;
    v2f A, B;
    A.x = (vaA != 0.0f && binA == m) ? 1.0f : 0.0f;
    A.y = (vaB != 0.0f && binB == m) ? 1.0f : 0.0f;
    B.x = isN0 ? cA * vaA : isN1 ? acA : isN2 ? vaA : 0.0f;
    B.y = isN0 ? cB * vaB : isN1 ? acB : isN2 ? vaB : 0.0f;
    acc = __builtin_amdgcn_wmma_f32_16x16x4_f32(
        false, A, false, B, (short)0, acc, false, false);
  };

  for (int t = 0; t < numIter; ++t) {
    const int base = t * rowsPerSweep + gw * 8 + s;
    int r0 = base, r1 = base + 4;
    int in0 = (r0 < N) ? 1 : 0, in1 = (r1 < N) ? 1 : 0;
    int rc0 = in0 ? r0 : N - 1, rc1 = in1 ? r1 : N - 1;

    // Coalesced: 8 lanes x 2 float4 = 256B per row; wave covers 2KB contiguous.
    const float4* rp0 = (const float4*)(probs + (size_t)rc0 * 64);
    const float4* rp1 = (const float4*)(probs + (size_t)rc1 * 64);
    float4 x0 = rp0[j], x1 = rp0[j + 8];
    float4 y0 = rp1[j], y1 = rp1[j + 8];
    int lab0 = labels[rc0] & 63, lab1 = labels[rc1] & 63;
    float pv0 = probs[(size_t)rc0 * 64 + lab0];   // WGP$ hit: line just fetched
    float pv1 = probs[(size_t)rc1 * 64 + lab1];

    // Branchless row max: local v_max3 tree + 3-step ds_swizzle xor butterfly.
    float b0 = fmaxf(fmaxf(fmaxf(x0.x, x0.y), fmaxf(x0.z, x0.w)),
                     fmaxf(fmaxf(x1.x, x1.y), fmaxf(x1.z, x1.w)));
    b0 = fmaxf(b0, swzf<0x041F>(b0));
    b0 = fmaxf(b0, swzf<0x081F>(b0));
    b0 = fmaxf(b0, swzf<0x101F>(b0));
    float b1 = fmaxf(fmaxf(fmaxf(y0.x, y0.y), fmaxf(y0.z, y0.w)),
                     fmaxf(fmaxf(y1.x, y1.y), fmaxf(y1.z, y1.w)));
    b1 = fmaxf(b1, swzf<0x041F>(b1));
    b1 = fmaxf(b1, swzf<0x081F>(b1));
    b1 = fmaxf(b1, swzf<0x101F>(b1));

    // accuracy == (prob at label == row max); exact argmax not needed
    int ac0 = (pv0 == b0) ? 1 : 0, ac1 = (pv1 == b1) ? 1 : 0;
    int va0 = (in0 && b0 > 0.0f) ? 1 : 0, va1 = (in1 && b1 > 0.0f) ? 1 : 0;
    // (lower, upper] binning: digitize(right=True)-1, clipped to [0,14]
    int n0 = (int)ceilf(b0 * 15.0f) - 1; n0 = n0 < 0 ? 0 : (n0 > 14 ? 14 : n0);
    int n1 = (int)ceilf(b1 * 15.0f) - 1; n1 = n1 < 0 ? 0 : (n1 > 14 ? 14 : n1);
    int code0 = n0 | (ac0 << 8) | (va0 << 9);
    int code1 = n1 | (ac1 << 8) | (va1 << 9);

    // Broadcast elements {0/2} (pattern 0x10: src=L&16) and {1/3} (0x110: src=(L&16)|8)
    accumulate(swzf<0x010>(b0), swzf<0x110>(b0), swzi<0x010>(code0), swzi<0x110>(code0));
    accumulate(swzf<0x010>(b1), swzf<0x110>(b1), swzi<0x010>(code1), swzi<0x110>(code1));
  }

  // D layout: VGPR v, lane L -> D[v + (L<16?0:8)][L%16]; cols 0..2 live.
  if (m < 3) {
    int mbase = (lane < 16) ? 0 : 8;
#pragma unroll
    for (int v = 0; v < 8; ++v)
      atomicAdd(&hist[m * 16 + mbase + v], acc[v]);   // ds_add_f32
  }
  __syncthreads();
  if (threadIdx.x < 48) atomicAdd(&ws[threadIdx.x], hist[threadIdx.x]);
}

__global__ void ece_finalize(const float* __restrict__ ws, float* __restrict__ out, int N) {
  if (threadIdx.x == 0) {
    float ece = 0.0f;
    for (int b = 0; b < ECE_NBINS; ++b) {
      float sc = ws[0 * 16 + b];
      float sa = ws[1 * 16 + b];
      float ct = ws[2 * 16 + b];
      if (ct > 0.0f) {
        float safe = fmaxf(ct, 1.0f);
        ece += fabsf(sc / safe - sa / safe) * (ct / (float)N);
      }
    }
    out[0] = ece;
  }
}

extern "C" void kernel_launch(void* const* d_in, const int* in_sizes, int n_in,
                              void* d_out, int out_size, void* d_ws, size_t ws_size,
                              hipStream_t stream) {
  (void)n_in; (void)out_size; (void)ws_size;
  const float* probs  = (const float*)d_in[0];
  const int*   labels = (const int*)d_in[1];
  float* ws  = (float*)d_ws;   // 48 floats: [n*16 + bin], n=0:conf 1:acc 2:count
  float* out = (float*)d_out;
  int N = in_sizes[1];         // number of rows (labels count)

  ece_zero_ws<<<1, 64, 0, stream>>>(ws);
  ece_main<<<ECE_BLOCKS, ECE_TPB, 0, stream>>>(probs, labels, ws, N);
  ece_finalize<<<1, 32, 0, stream>>>(ws, out, N);
}